// GraphSAGENet_73581379715727
// MI455X (gfx1250) — compile-verified
//
#include <hip/hip_runtime.h>
#include <hip/hip_bf16.h>

#define N_NODES  100000
#define N_EDGES  400000
#define N_GRAPHS 5000
#define IN_DIM   11
#define HID      128
#define BN_EPS   1e-5f

typedef __attribute__((ext_vector_type(2))) float v2f;
typedef __attribute__((ext_vector_type(8))) float v8f;

// ---------------------------------------------------------------------------
// Elementwise / irregular (bandwidth-bound) kernels
// ---------------------------------------------------------------------------

__global__ void k_zero(float* __restrict__ p, long n) {
  long i = (long)blockIdx.x * blockDim.x + threadIdx.x;
  long stride = (long)gridDim.x * blockDim.x;
  for (; i < n; i += stride) p[i] = 0.0f;
}

__global__ void k_deg(const int* __restrict__ tgt, float* __restrict__ deg) {
  int e = blockIdx.x * blockDim.x + threadIdx.x;
  if (e < N_EDGES) atomicAdd(&deg[tgt[e]], 1.0f);
}

__global__ void k_invdeg(float* __restrict__ deg) {
  int i = blockIdx.x * blockDim.x + threadIdx.x;
  if (i < N_NODES) deg[i] = 1.0f / fmaxf(deg[i], 1.0f);
}

// scatter-add of 11-dim input features (layer 0 aggregation)
__global__ void k_scatter0(const float* __restrict__ x, const int* __restrict__ src,
                           const int* __restrict__ tgt, float* __restrict__ aggr) {
  int e = blockIdx.x * blockDim.x + threadIdx.x;
  if (e >= N_EDGES) return;
  long s = (long)src[e] * IN_DIM;
  long d = (long)tgt[e] * IN_DIM;
#pragma unroll
  for (int k = 0; k < IN_DIM; ++k) atomicAdd(&aggr[d + k], x[s + k]);
}

// scatter-add of 128-dim hidden features: one wave per edge, 4 feats per lane
__global__ void k_scatter128(const float* __restrict__ h, const int* __restrict__ src,
                             const int* __restrict__ tgt, float* __restrict__ aggr) {
  long t = (long)blockIdx.x * blockDim.x + threadIdx.x;
  long e = t >> 5;
  if (e >= N_EDGES) return;
  int lane = (int)(t & 31);
  long s = (long)src[e] * HID + lane * 4;
  long d = (long)tgt[e] * HID + lane * 4;
  float4 v = *(const float4*)(h + s);
  atomicAdd(&aggr[d + 0], v.x);
  atomicAdd(&aggr[d + 1], v.y);
  atomicAdd(&aggr[d + 2], v.z);
  atomicAdd(&aggr[d + 3], v.w);
}

// layer 0: K=11 too skinny for WMMA -> fused VALU dot + bias + BN + ReLU
__global__ void k_layer0(const float* __restrict__ x, const float* __restrict__ aggr11,
                         const float* __restrict__ inv_deg,
                         const float* __restrict__ Wl, const float* __restrict__ Wr,
                         const float* __restrict__ bias,
                         const float* __restrict__ gamma, const float* __restrict__ beta,
                         const float* __restrict__ mean, const float* __restrict__ var,
                         float* __restrict__ h_out) {
  long t = (long)blockIdx.x * blockDim.x + threadIdx.x;
  if (t >= (long)N_NODES * HID) return;
  int n = (int)(t >> 7);
  int f = (int)(t & 127);
  float id = inv_deg[n];
  float v = bias[f];
#pragma unroll
  for (int d = 0; d < IN_DIM; ++d) {
    v += aggr11[(long)n * IN_DIM + d] * id * Wl[d * HID + f];
    v += x[(long)n * IN_DIM + d] * Wr[d * HID + f];
  }
  v = (v - mean[f]) * rsqrtf(var[f] + BN_EPS) * gamma[f] + beta[f];
  h_out[t] = fmaxf(v, 0.0f);
}

__global__ void k_counts(const int* __restrict__ batch, float* __restrict__ cnt) {
  int n = blockIdx.x * blockDim.x + threadIdx.x;
  if (n < N_NODES) atomicAdd(&cnt[batch[n]], 1.0f);
}

__global__ void k_invcnt(float* __restrict__ cnt) {
  int g = blockIdx.x * blockDim.x + threadIdx.x;
  if (g < N_GRAPHS) cnt[g] = 1.0f / fmaxf(cnt[g], 1.0f);
}

// global mean-pool (sum part): one wave per node, 4 feats per lane
__global__ void k_pool(const float* __restrict__ h, const int* __restrict__ batch,
                       float* __restrict__ pooled) {
  long t = (long)blockIdx.x * blockDim.x + threadIdx.x;
  long n = t >> 5;
  if (n >= N_NODES) return;
  int lane = (int)(t & 31);
  int b = batch[n];
  float4 v = *(const float4*)(h + n * HID + lane * 4);
  float* p = pooled + (long)b * HID + lane * 4;
  atomicAdd(p + 0, v.x);
  atomicAdd(p + 1, v.y);
  atomicAdd(p + 2, v.z);
  atomicAdd(p + 3, v.w);
}

// final 128 -> 1 projection
__global__ void k_head(const float* __restrict__ z, const float* __restrict__ W3,
                       const float* __restrict__ b3, float* __restrict__ out) {
  int g = blockIdx.x * blockDim.x + threadIdx.x;
  if (g >= N_GRAPHS) return;
  float v = b3[0];
#pragma unroll 4
  for (int k = 0; k < HID; ++k) v += z[(long)g * HID + k] * W3[k];
  out[g] = v;
}

// ---------------------------------------------------------------------------
// WMMA f32 16x16x4 building blocks
//   A 16x4 f32: lanes 0-15 row M=lane hold K+0/K+1; lanes 16-31 hold K+2/K+3.
//   B 4x16 f32: mirrors C striping (N = lane&15; halves carry K pairs).
//   C/D 16x16 f32: VGPR r -> M=r (lanes 0-15) / M=r+8 (lanes 16-31), N=lane&15.
// ---------------------------------------------------------------------------

// Single M-tile: acc += A_lds(16x128, row stride 128) @ W[:, colBase:+16]
__device__ __forceinline__ v8f wmma_ldsA_times_W(const float* As,
                                                 const float* __restrict__ W,
                                                 int colBase, v8f acc) {
  const int lane = threadIdx.x & 31;
  const int half = lane >> 4;
  const int l16  = lane & 15;
  const float* Wcol = W + colBase + l16;
#pragma unroll
  for (int k = 0; k < HID; k += 4) {
    const int ka = k + 2 * half;
    v2f a = *(const v2f*)&As[l16 * HID + ka];       // ka even -> 8B aligned
    v2f b;
    b.x = Wcol[ka * HID];
    b.y = Wcol[(ka + 1) * HID];
    acc = __builtin_amdgcn_wmma_f32_16x16x4_f32(false, a, false, b,
                                                (short)0, acc, false, false);
  }
  return acc;
}

// Dual M-tile: one B fragment feeds two WMMAs (rows [0,16) and [16,32) of tile)
__device__ __forceinline__ void wmma_ldsA2_times_W(const float* As,
                                                   const float* __restrict__ W,
                                                   int colBase, v8f* acc0, v8f* acc1) {
  const int lane = threadIdx.x & 31;
  const int half = lane >> 4;
  const int l16  = lane & 15;
  const float* Wcol = W + colBase + l16;
  v8f a0c = *acc0, a1c = *acc1;
#pragma unroll
  for (int k = 0; k < HID; k += 4) {
    const int ka = k + 2 * half;
    v2f a0 = *(const v2f*)&As[l16 * HID + ka];
    v2f a1 = *(const v2f*)&As[(l16 + 16) * HID + ka];
    v2f b;
    b.x = Wcol[ka * HID];
    b.y = Wcol[(ka + 1) * HID];
    a0c = __builtin_amdgcn_wmma_f32_16x16x4_f32(false, a0, false, b,
                                                (short)0, a0c, false, false);
    a1c = __builtin_amdgcn_wmma_f32_16x16x4_f32(false, a1, false, b,
                                                (short)0, a1c, false, false);
  }
  *acc0 = a0c;
  *acc1 = a1c;
}

// Fused SAGE layer (dims 128->128):
//   h_out = relu(BN((aggr*inv_deg)@Wl + h_in@Wr + b)) + h_in
// 256 threads = 8 waves; block owns 32 nodes; wave w owns columns [16w,16w+16)
// with two 16-row accumulators sharing each B fragment.
// h_out may alias aggr: each block stages its aggr rows into LDS before storing.
__launch_bounds__(256)
__global__ void k_sage(const float* __restrict__ aggr, const float* __restrict__ h_in,
                       const float* __restrict__ inv_deg,
                       const float* __restrict__ Wl, const float* __restrict__ Wr,
                       const float* __restrict__ bias,
                       const float* __restrict__ gamma, const float* __restrict__ beta,
                       const float* __restrict__ mean, const float* __restrict__ var,
                       float* __restrict__ h_out) {
  __shared__ float As[32 * HID];
  __shared__ float Hs[32 * HID];
  const int nodeBase = blockIdx.x * 32;

  // Warm L2 for the weight panels this block will stream (global_prefetch_b8).
  if (threadIdx.x < HID) {
    __builtin_prefetch(Wl + threadIdx.x * HID, 0, 2);
    __builtin_prefetch(Wr + threadIdx.x * HID, 0, 2);
  }

  for (int i = threadIdx.x; i < 32 * HID; i += 256) {
    int r = i >> 7;
    long g = (long)(nodeBase + r) * HID + (i & 127);
    As[i] = aggr[g] * inv_deg[nodeBase + r];
    Hs[i] = h_in[g];
  }
  __syncthreads();

  const int colBase = (threadIdx.x >> 5) * 16;
  v8f acc0 = {}, acc1 = {};
  wmma_ldsA2_times_W(As, Wl, colBase, &acc0, &acc1);
  wmma_ldsA2_times_W(Hs, Wr, colBase, &acc0, &acc1);

  const int lane = threadIdx.x & 31;
  const int half = lane >> 4;
  const int f = colBase + (lane & 15);
  const float bi = bias[f];
  const float sc = rsqrtf(var[f] + BN_EPS) * gamma[f];
  const float mu = mean[f];
  const float be = beta[f];
#pragma unroll
  for (int r = 0; r < 8; ++r) {
    int m0 = r + 8 * half;        // rows [0,16)
    int m1 = m0 + 16;             // rows [16,32)
    float v0 = (acc0[r] + bi - mu) * sc + be;
    float v1 = (acc1[r] + bi - mu) * sc + be;
    v0 = fmaxf(v0, 0.0f) + Hs[m0 * HID + f];   // residual
    v1 = fmaxf(v1, 0.0f) + Hs[m1 * HID + f];
    h_out[(long)(nodeBase + m0) * HID + f] = v0;
    h_out[(long)(nodeBase + m1) * HID + f] = v1;
  }
}

// Generic fused dense layer: out = [relu](rowscale(A) @ W + b), A is Mx128
__launch_bounds__(256)
__global__ void k_mlp(const float* __restrict__ A, const float* __restrict__ row_scale,
                      const float* __restrict__ W, const float* __restrict__ bias,
                      float* __restrict__ out, int M, int do_relu) {
  __shared__ float As[16 * HID];
  const int rowBase = blockIdx.x * 16;
  for (int i = threadIdx.x; i < 16 * HID; i += 256) {
    int r = i >> 7;
    int row = rowBase + r;
    float v = 0.0f;
    if (row < M) {
      v = A[(long)row * HID + (i & 127)];
      if (row_scale) v *= row_scale[row];
    }
    As[i] = v;
  }
  __syncthreads();

  const int colBase = (threadIdx.x >> 5) * 16;
  v8f acc = {};
  acc = wmma_ldsA_times_W(As, W, colBase, acc);

  const int lane = threadIdx.x & 31;
  const int half = lane >> 4;
  const int f = colBase + (lane & 15);
  const float bi = bias[f];
#pragma unroll
  for (int r = 0; r < 8; ++r) {
    int row = rowBase + r + 8 * half;
    if (row < M) {
      float v = acc[r] + bi;
      if (do_relu) v = fmaxf(v, 0.0f);
      out[(long)row * HID + f] = v;
    }
  }
}

// ---------------------------------------------------------------------------
// Host orchestration
// ---------------------------------------------------------------------------
extern "C" void kernel_launch(void* const* d_in, const int* in_sizes, int n_in,
                              void* d_out, int out_size, void* d_ws, size_t ws_size,
                              hipStream_t stream) {
  (void)in_sizes; (void)n_in; (void)out_size; (void)ws_size;

  const float* x        = (const float*)d_in[0];
  const int*   ei       = (const int*)d_in[1];
  const int*   batch    = (const int*)d_in[2];
  const float* c0_Wl    = (const float*)d_in[3];
  const float* c0_Wr    = (const float*)d_in[4];
  const float* c0_b     = (const float*)d_in[5];
  const float* cs_Wl    = (const float*)d_in[6];   // [3][128][128]
  const float* cs_Wr    = (const float*)d_in[7];
  const float* cs_b     = (const float*)d_in[8];   // [3][128]
  const float* bn_g     = (const float*)d_in[9];   // [4][128]
  const float* bn_b     = (const float*)d_in[10];
  const float* bn_m     = (const float*)d_in[11];
  const float* bn_v     = (const float*)d_in[12];
  const float* W1       = (const float*)d_in[13];
  const float* b1       = (const float*)d_in[14];
  const float* W2       = (const float*)d_in[15];
  const float* b2       = (const float*)d_in[16];
  const float* W3       = (const float*)d_in[17];
  const float* b3       = (const float*)d_in[18];
  float*       out      = (float*)d_out;

  const int* src = ei;
  const int* tgt = ei + N_EDGES;

  // Workspace carve-up (floats)
  float* ws      = (float*)d_ws;
  const long NH  = (long)N_NODES * HID;          // 12.8M
  float* bufA    = ws;                            // node features ping
  float* bufB    = bufA + NH;                     // node features pong (aggr/h alias)
  float* inv_deg = bufB + NH;                     // [N_NODES]
  float* pooled  = inv_deg + N_NODES;             // [N_GRAPHS*HID]
  float* invcnt  = pooled + (long)N_GRAPHS * HID; // [N_GRAPHS]
  float* z1      = invcnt + N_GRAPHS;             // [N_GRAPHS*HID]
  float* z2      = z1 + (long)N_GRAPHS * HID;     // [N_GRAPHS*HID]

  const int TB = 256;
  auto blocks = [](long n, int tb) { return (int)((n + tb - 1) / tb); };

  // --- degree / inverse degree ---
  k_zero<<<blocks(N_NODES, TB), TB, 0, stream>>>(inv_deg, N_NODES);
  k_deg<<<blocks(N_EDGES, TB), TB, 0, stream>>>(tgt, inv_deg);
  k_invdeg<<<blocks(N_NODES, TB), TB, 0, stream>>>(inv_deg);

  // --- layer 0 (11 -> 128) ---
  k_zero<<<blocks((long)N_NODES * IN_DIM, TB), TB, 0, stream>>>(bufB, (long)N_NODES * IN_DIM);
  k_scatter0<<<blocks(N_EDGES, TB), TB, 0, stream>>>(x, src, tgt, bufB);
  k_layer0<<<blocks(NH, TB), TB, 0, stream>>>(x, bufB, inv_deg, c0_Wl, c0_Wr, c0_b,
                                              bn_g, bn_b, bn_m, bn_v, bufA);

  // --- layers 1..3 (128 -> 128), WMMA; ping-pong bufA/bufB ---
  float* hc    = bufA;
  float* other = bufB;
  for (int i = 0; i < 3; ++i) {
    k_zero<<<blocks(NH, TB), TB, 0, stream>>>(other, NH);
    k_scatter128<<<blocks((long)N_EDGES * 32, TB), TB, 0, stream>>>(hc, src, tgt, other);
    k_sage<<<(N_NODES + 31) / 32, TB, 0, stream>>>(
        other /*aggr*/, hc, inv_deg,
        cs_Wl + (long)i * HID * HID, cs_Wr + (long)i * HID * HID, cs_b + i * HID,
        bn_g + (i + 1) * HID, bn_b + (i + 1) * HID,
        bn_m + (i + 1) * HID, bn_v + (i + 1) * HID,
        other /*h_out aliases aggr: safe, rows staged to LDS first*/);
    float* t = hc; hc = other; other = t;
  }

  // --- global mean pool ---
  k_zero<<<blocks((long)N_GRAPHS * HID, TB), TB, 0, stream>>>(pooled, (long)N_GRAPHS * HID);
  k_zero<<<blocks(N_GRAPHS, TB), TB, 0, stream>>>(invcnt, N_GRAPHS);
  k_counts<<<blocks(N_NODES, TB), TB, 0, stream>>>(batch, invcnt);
  k_pool<<<blocks((long)N_NODES * 32, TB), TB, 0, stream>>>(hc, batch, pooled);
  k_invcnt<<<blocks(N_GRAPHS, TB), TB, 0, stream>>>(invcnt);

  // --- MLP head (WMMA), mean-division folded into row_scale of first layer ---
  const int mlpBlocks = (N_GRAPHS + 15) / 16;
  k_mlp<<<mlpBlocks, TB, 0, stream>>>(pooled, invcnt, W1, b1, z1, N_GRAPHS, 1);
  k_mlp<<<mlpBlocks, TB, 0, stream>>>(z1, nullptr, W2, b2, z2, N_GRAPHS, 1);
  k_head<<<blocks(N_GRAPHS, TB), TB, 0, stream>>>(z2, W3, b3, out);
}